// EvolvedLoopLinear_16363825398479
// MI455X (gfx1250) — compile-verified
//
#include <hip/hip_runtime.h>
#include <hip/hip_bf16.h>

typedef __attribute__((ext_vector_type(2))) float v2f;
typedef __attribute__((ext_vector_type(4))) float f4v;
typedef __attribute__((ext_vector_type(8))) float v8f;

#define IN_F   8192
#define OUT_F  4096
#define NGROUP 128
#define STEP   64      // K per group
#define MPG    32      // outputs per group
#define ROWS   128     // batch rows per workgroup
#define PAD    68      // padded LDS row stride in floats (bank-conflict-free)

__global__ __launch_bounds__(256)
void grouped_linear_wmma_f32(const float* __restrict__ x,
                             const float* __restrict__ weight,
                             const float* __restrict__ bias,
                             float* __restrict__ out)
{
    __shared__ float Xs[ROWS * PAD];   // 128 x 64 (padded)  ~34.8 KB
    __shared__ float Ws[MPG  * PAD];   //  32 x 64 (padded)  ~8.7 KB

    const int g  = blockIdx.x;          // group id 0..127  (fastest-varying:
                                        //  all groups of a row tile run together
                                        //  so output lines combine in L2 quickly)
    const int tb = blockIdx.y * ROWS;   // batch-row tile base
    const int t  = threadIdx.x;         // 0..255

    // ---- stage X tile: 128 rows x 64 floats, float4 coalesced, non-temporal
    //      (X is streamed exactly once; keep it out of L2) ----
    const size_t xbase = (size_t)tb * IN_F + (size_t)g * STEP;
    #pragma unroll
    for (int i = 0; i < 8; ++i) {
        int flat = i * 256 + t;                 // 0..2047  (2048 float4s)
        int r    = flat >> 4;                   // row 0..127 (16 float4 per row)
        int c4   = flat & 15;
        f4v v = __builtin_nontemporal_load(
                    (const f4v*)(x + xbase + (size_t)r * IN_F + c4 * 4));
        *(f4v*)(&Xs[r * PAD + c4 * 4]) = v;
    }
    // ---- stage W tile: 32 rows (j = m*128+g) x 64 floats (reused: regular) ----
    #pragma unroll
    for (int i = 0; i < 2; ++i) {
        int flat = i * 256 + t;                 // 0..511   (512 float4s)
        int m    = flat >> 4;                   // 0..31
        int c4   = flat & 15;
        size_t wrow = (size_t)(m * NGROUP + g) * IN_F + (size_t)g * STEP;
        f4v v = *(const f4v*)(weight + wrow + c4 * 4);
        *(f4v*)(&Ws[m * PAD + c4 * 4]) = v;
    }
    __syncthreads();

    // ---- WMMA compute: each wave = 16-row strip x both 16-col m-tiles ----
    const int w  = t >> 5;       // wave 0..7
    const int l  = t & 31;       // lane
    const int lo = l & 15;
    const int hi = l >> 4;       // 0 or 1

    // A fragment: lane holds row (w*16+lo), K = k0 + 2*hi + {0,1}
    const float* xrow = &Xs[(w * 16 + lo) * PAD + 2 * hi];
    // B fragments: lane holds col n=lo (m-tile 0/1), same K pattern
    const float* wr0  = &Ws[lo * PAD + 2 * hi];
    const float* wr1  = &Ws[(16 + lo) * PAD + 2 * hi];

    v8f c0 = {};
    v8f c1 = {};
    #pragma unroll
    for (int k0 = 0; k0 < STEP; k0 += 4) {
        v2f a  = *(const v2f*)(xrow + k0);
        v2f b0 = *(const v2f*)(wr0 + k0);
        v2f b1 = *(const v2f*)(wr1 + k0);
        c0 = __builtin_amdgcn_wmma_f32_16x16x4_f32(
                 /*neg_a=*/false, a, /*neg_b=*/false, b0,
                 /*c_mod=*/(short)0, c0, /*reuse_a=*/false, /*reuse_b=*/false);
        c1 = __builtin_amdgcn_wmma_f32_16x16x4_f32(
                 /*neg_a=*/false, a, /*neg_b=*/false, b1,
                 /*c_mod=*/(short)0, c1, /*reuse_a=*/false, /*reuse_b=*/false);
    }

    // ---- bias + store: D[v][lo] -> out[row = tb + w*16 + v + 8*hi][j] ----
    const int j0  = lo * NGROUP + g;          // m-tile 0:  m = lo
    const int j1  = (16 + lo) * NGROUP + g;   // m-tile 1:  m = 16 + lo
    const float b0v = bias[j0];
    const float b1v = bias[j1];
    const int rbase = tb + w * 16 + 8 * hi;
    #pragma unroll
    for (int v = 0; v < 8; ++v) {
        size_t ro = (size_t)(rbase + v) * OUT_F;
        out[ro + j0] = c0[v] + b0v;
        out[ro + j1] = c1[v] + b1v;
    }
}

extern "C" void kernel_launch(void* const* d_in, const int* in_sizes, int n_in,
                              void* d_out, int out_size, void* d_ws, size_t ws_size,
                              hipStream_t stream) {
    const float* x      = (const float*)d_in[0];
    const float* weight = (const float*)d_in[1];
    const float* bias   = (const float*)d_in[2];
    float* out          = (float*)d_out;

    dim3 grid(NGROUP, 4096 / ROWS);   // (128 groups fastest, 32 row tiles)
    dim3 block(256);
    grouped_linear_wmma_f32<<<grid, block, 0, stream>>>(x, weight, bias, out);
}